// NonLocalBlock_78271484003083
// MI455X (gfx1250) — compile-verified
//
#include <hip/hip_runtime.h>
#include <hip/hip_bf16.h>

// NonLocalBlock, algebraically collapsed (no softmax => attention is bilinear):
//   out = x + A2[b] @ X~[b],  A2 built from Gram matrix G~ = X~ X~^T per batch.
// Heavy GEMMs use V_WMMA_F32_16X16X4_F32 (full fp32, matches reference).
//
// WMMA f32 16x16x4 fragment layout (CDNA5 ISA 7.12.2):
//   A (16x4): lane L holds A[m][k], m = L&15, k = 2*(L>>4) + vgpr  -> float2 @ row m, col 2*(L>>4)
//   B (4x16): lane L holds B[k][n], n = L&15, k = 2*(L>>4) + vgpr
//   C/D (16x16): vgpr v, lane L -> row m = v + 8*(L>>4), col n = L&15

typedef __attribute__((ext_vector_type(2))) float v2f;
typedef __attribute__((ext_vector_type(8))) float v8f;

#define WMMA_F32(a, b, c) \
    __builtin_amdgcn_wmma_f32_16x16x4_f32(false, (a), false, (b), (short)0, (c), false, false)

static constexpr int B_  = 16;
static constexpr int C_  = 256;
static constexpr int IC_ = 128;
static constexpr int HW_ = 4096;
static constexpr int CP_ = 272;   // 256 channels + ones row, padded to 17*16
static constexpr int KCH = 4;     // K-split chunks for the Gram kernel

// ---------------------------------------------------------------------------
// Augmented weights: Wph_aug/Wg_aug [128][272] = [W | bias | 0],
// WthT [272][128] = [W_theta^T ; b_theta ; 0]
__global__ void prep_weights(const float* __restrict__ w_theta, const float* __restrict__ b_theta,
                             const float* __restrict__ w_phi,   const float* __restrict__ b_phi,
                             const float* __restrict__ w_g,     const float* __restrict__ b_g,
                             float* __restrict__ wph_aug, float* __restrict__ wg_aug,
                             float* __restrict__ wthT) {
  int tid = blockIdx.x * blockDim.x + threadIdx.x;
  if (tid >= IC_ * CP_) return;                       // 128*272 == 272*128
  int i = tid / CP_, j = tid % CP_;
  float vp = 0.f, vg = 0.f;
  if (j < C_)        { vp = w_phi[i * C_ + j]; vg = w_g[i * C_ + j]; }
  else if (j == C_)  { vp = b_phi[i];          vg = b_g[i]; }
  wph_aug[tid] = vp;
  wg_aug[tid]  = vg;
  int r = tid / IC_, c = tid % IC_;
  float vt = 0.f;
  if (r < C_)        vt = w_theta[c * C_ + r];
  else if (r == C_)  vt = b_theta[c];
  wthT[tid] = vt;
}

// ---------------------------------------------------------------------------
// s[b][c] = sum_hw x[b][c][hw]   (one wave per row, shfl reduction)
__global__ void rowsum(const float* __restrict__ x, float* __restrict__ s) {
  int row = blockIdx.x, b = blockIdx.y, lane = threadIdx.x;
  const float* p = x + ((size_t)b * C_ + row) * HW_;
  float acc = 0.f;
  for (int i = lane; i < HW_; i += 32) acc += p[i];
  #pragma unroll
  for (int off = 16; off > 0; off >>= 1) acc += __shfl_down(acc, off, 32);
  if (lane == 0) s[b * C_ + row] = acc;
}

// ---------------------------------------------------------------------------
// Border of Gt (rows/cols >= 256): row/col 256 = rowsums, corner = HW, rest 0.
__global__ void fill_border(const float* __restrict__ s, float* __restrict__ Gt) {
  int tid = blockIdx.x * blockDim.x + threadIdx.x;
  if (tid >= B_ * CP_ * CP_) return;
  int j = tid % CP_, i = (tid / CP_) % CP_, b = tid / (CP_ * CP_);
  if (i < C_ && j < C_) return;                 // bulk written by reduce_gram
  float v = 0.f;
  if (i == C_ && j == C_)      v = (float)HW_;
  else if (i == C_ && j < C_)  v = s[b * C_ + j];
  else if (j == C_ && i < C_)  v = s[b * C_ + i];
  Gt[tid] = v;
}

// ---------------------------------------------------------------------------
// Partial Gram over K chunk: Gp[ch][b][256][256] = X[:, kbeg:kend] X[:, kbeg:kend]^T
// One wave computes a 32x32 tile (2x2 WMMA blocking), branch-free inner loop.
__global__ void gram_partial(const float* __restrict__ x, float* __restrict__ Gp) {
  int z = blockIdx.z, b = z / KCH, ch = z % KCH;
  int m0 = blockIdx.y * 32, n0 = blockIdx.x * 32;
  int lane = threadIdx.x, r = lane & 15, kq = lane >> 4;
  const float* xb  = x + (size_t)b * C_ * HW_;
  const float* pa0 = xb + (size_t)(m0 + r) * HW_;
  const float* pa1 = pa0 + (size_t)16 * HW_;
  const float* pb0 = xb + (size_t)(n0 + r) * HW_;
  const float* pb1 = pb0 + (size_t)16 * HW_;
  int kbeg = ch * (HW_ / KCH), kend = kbeg + (HW_ / KCH);
  v8f acc00 = {}, acc01 = {}, acc10 = {}, acc11 = {};
  for (int k0 = kbeg; k0 < kend; k0 += 4) {
    int k = k0 + 2 * kq;
    v2f a0 = *(const v2f*)(pa0 + k);
    v2f a1 = *(const v2f*)(pa1 + k);
    v2f b0 = *(const v2f*)(pb0 + k);
    v2f b1 = *(const v2f*)(pb1 + k);
    acc00 = WMMA_F32(a0, b0, acc00);
    acc01 = WMMA_F32(a0, b1, acc01);
    acc10 = WMMA_F32(a1, b0, acc10);
    acc11 = WMMA_F32(a1, b1, acc11);
  }
  float* gp = Gp + ((size_t)ch * B_ + b) * C_ * C_;
  #pragma unroll
  for (int v = 0; v < 8; ++v) {
    int m = v + 8 * kq;
    gp[(size_t)(m0 + m) * C_ + n0 + r]           = acc00[v];
    gp[(size_t)(m0 + m) * C_ + n0 + 16 + r]      = acc01[v];
    gp[(size_t)(m0 + 16 + m) * C_ + n0 + r]      = acc10[v];
    gp[(size_t)(m0 + 16 + m) * C_ + n0 + 16 + r] = acc11[v];
  }
}

// ---------------------------------------------------------------------------
// Gt bulk = sum of K chunks (deterministic fixed-order reduction).
__global__ void reduce_gram(const float* __restrict__ Gp, float* __restrict__ Gt) {
  int tid = blockIdx.x * blockDim.x + threadIdx.x;
  if (tid >= B_ * C_ * C_) return;
  int j = tid % C_, i = (tid / C_) % C_, b = tid / (C_ * C_);
  float v = 0.f;
  #pragma unroll
  for (int ch = 0; ch < KCH; ++ch)
    v += Gp[((size_t)ch * B_ + b) * C_ * C_ + (size_t)i * C_ + j];
  Gt[((size_t)b * CP_ + i) * CP_ + j] = v;
}

// ---------------------------------------------------------------------------
// Generic batched C[M,N] = alpha * A[M,K] * B[N,K]^T (row-major, K%4==0, M/N%16==0).
__global__ void gemm_nt(const float* __restrict__ A, long sA, int lda,
                        const float* __restrict__ B, long sB, int ldb,
                        float* __restrict__ C, long sC, int ldc,
                        int K, float alpha) {
  int b  = blockIdx.z;
  int m0 = blockIdx.y * 16, n0 = blockIdx.x * 16;
  int lane = threadIdx.x, r = lane & 15, kq = lane >> 4;
  const float* pa = A + (size_t)b * sA + (size_t)(m0 + r) * lda;
  const float* pb = B + (size_t)b * sB + (size_t)(n0 + r) * ldb;
  v8f acc = {};
  for (int k0 = 0; k0 < K; k0 += 4) {
    v2f a  = *(const v2f*)(pa + k0 + 2 * kq);
    v2f bb = *(const v2f*)(pb + k0 + 2 * kq);
    acc = WMMA_F32(a, bb, acc);
  }
  float* pc = C + (size_t)b * sC;
  #pragma unroll
  for (int v = 0; v < 8; ++v) {
    int m = m0 + v + 8 * kq;
    pc[(size_t)m * ldc + n0 + r] = alpha * acc[v];
  }
}

// ---------------------------------------------------------------------------
// A2[b][c][j] = scale[c]*Atil[b][c][j]  (+ scale*b_out + shift at j==256)
__global__ void finalize_A(const float* __restrict__ Atil, const float* __restrict__ b_out,
                           const float* __restrict__ gamma, const float* __restrict__ beta,
                           const float* __restrict__ mean,  const float* __restrict__ var,
                           float* __restrict__ A2) {
  long tid = (long)blockIdx.x * blockDim.x + threadIdx.x;
  if (tid >= (long)B_ * C_ * CP_) return;
  int j = (int)(tid % CP_);
  int c = (int)((tid / CP_) % C_);
  float scale = gamma[c] * rsqrtf(var[c] + 1e-5f);
  float v = scale * Atil[tid];
  if (j == C_) v += scale * b_out[c] + (beta[c] - mean[c] * scale);
  A2[tid] = v;
}

// ---------------------------------------------------------------------------
// out[b] = x[b] + A2[b][:, :256] @ x[b] + A2[b][:, 256]  (2x2 WMMA blocking)
__global__ void final_out(const float* __restrict__ x, const float* __restrict__ A2,
                          float* __restrict__ out) {
  int b  = blockIdx.z;
  int m0 = blockIdx.y * 32;   // output channel tile
  int n0 = blockIdx.x * 32;   // spatial tile
  int lane = threadIdx.x, r = lane & 15, kq = lane >> 4;
  const float* a2b = A2 + (size_t)b * C_ * CP_;
  const float* pa0 = a2b + (size_t)(m0 + r) * CP_;
  const float* pa1 = pa0 + (size_t)16 * CP_;
  const float* xb  = x + (size_t)b * C_ * HW_;
  v8f acc00 = {}, acc01 = {}, acc10 = {}, acc11 = {};
  for (int k0 = 0; k0 < C_; k0 += 4) {
    int k = k0 + 2 * kq;
    v2f a0 = *(const v2f*)(pa0 + k);
    v2f a1 = *(const v2f*)(pa1 + k);
    v2f b0, b1;
    b0[0] = xb[(size_t)k * HW_ + n0 + r];
    b0[1] = xb[(size_t)(k + 1) * HW_ + n0 + r];
    b1[0] = xb[(size_t)k * HW_ + n0 + 16 + r];
    b1[1] = xb[(size_t)(k + 1) * HW_ + n0 + 16 + r];
    acc00 = WMMA_F32(a0, b0, acc00);
    acc01 = WMMA_F32(a0, b1, acc01);
    acc10 = WMMA_F32(a1, b0, acc10);
    acc11 = WMMA_F32(a1, b1, acc11);
  }
  #pragma unroll
  for (int v = 0; v < 8; ++v) {
    int m  = v + 8 * kq;
    int c0 = m0 + m, c1 = m0 + 16 + m;
    float bias0 = a2b[(size_t)c0 * CP_ + C_];
    float bias1 = a2b[(size_t)c1 * CP_ + C_];
    size_t i00 = (size_t)b * C_ * HW_ + (size_t)c0 * HW_ + n0 + r;
    size_t i10 = (size_t)b * C_ * HW_ + (size_t)c1 * HW_ + n0 + r;
    out[i00]      = x[i00]      + acc00[v] + bias0;
    out[i00 + 16] = x[i00 + 16] + acc01[v] + bias0;
    out[i10]      = x[i10]      + acc10[v] + bias1;
    out[i10 + 16] = x[i10 + 16] + acc11[v] + bias1;
  }
}

// ---------------------------------------------------------------------------
extern "C" void kernel_launch(void* const* d_in, const int* in_sizes, int n_in,
                              void* d_out, int out_size, void* d_ws, size_t ws_size,
                              hipStream_t stream) {
  const float* x       = (const float*)d_in[0];
  const float* w_theta = (const float*)d_in[1];
  const float* b_theta = (const float*)d_in[2];
  const float* w_phi   = (const float*)d_in[3];
  const float* b_phi   = (const float*)d_in[4];
  const float* w_g     = (const float*)d_in[5];
  const float* b_g     = (const float*)d_in[6];
  const float* w_out   = (const float*)d_in[7];
  const float* b_out   = (const float*)d_in[8];
  const float* bn_gamma= (const float*)d_in[9];
  const float* bn_beta = (const float*)d_in[10];
  const float* bn_mean = (const float*)d_in[11];
  const float* bn_var  = (const float*)d_in[12];
  float* out = (float*)d_out;

  float* ws = (float*)d_ws;
  size_t o = 0;
  float* Gt   = ws + o; o += (size_t)B_ * CP_ * CP_;
  float* Gp   = ws + o; o += (size_t)KCH * B_ * C_ * C_;
  float* sums = ws + o; o += (size_t)B_ * C_;
  float* wph  = ws + o; o += (size_t)IC_ * CP_;
  float* wg   = ws + o; o += (size_t)IC_ * CP_;
  float* wthT = ws + o; o += (size_t)CP_ * IC_;
  float* U    = ws + o; o += (size_t)B_ * IC_ * CP_;
  float* Mt   = ws + o; o += (size_t)B_ * IC_ * IC_;
  float* T2   = ws + o; o += (size_t)B_ * C_ * IC_;
  float* Atil = ws + o; o += (size_t)B_ * C_ * CP_;
  float* A2   = ws + o; o += (size_t)B_ * C_ * CP_;
  if (ws_size < o * sizeof(float)) return;

  // 1. augmented weights
  prep_weights<<<(IC_ * CP_ + 255) / 256, 256, 0, stream>>>(
      w_theta, b_theta, w_phi, b_phi, w_g, b_g, wph, wg, wthT);

  // 2. row sums (-> Gram border) and Gram bulk partials
  rowsum<<<dim3(C_, B_), 32, 0, stream>>>(x, sums);
  fill_border<<<(B_ * CP_ * CP_ + 255) / 256, 256, 0, stream>>>(sums, Gt);
  gram_partial<<<dim3(C_ / 32, C_ / 32, B_ * KCH), 32, 0, stream>>>(x, Gp);
  reduce_gram<<<(B_ * C_ * C_ + 255) / 256, 256, 0, stream>>>(Gp, Gt);

  // 3. U = Wph_aug * Gt^T (Gt symmetric) -> [B,128,272], K=272
  gemm_nt<<<dim3(CP_ / 16, IC_ / 16, B_), 32, 0, stream>>>(
      wph, 0, CP_, Gt, (long)CP_ * CP_, CP_, U, (long)IC_ * CP_, CP_, CP_, 1.0f);

  // 4. Mt = U * Wg_aug^T = M^T -> [B,128,128], K=272
  gemm_nt<<<dim3(IC_ / 16, IC_ / 16, B_), 32, 0, stream>>>(
      U, (long)IC_ * CP_, CP_, wg, 0, CP_, Mt, (long)IC_ * IC_, IC_, CP_, 1.0f);

  // 5. T2 = W_out * Mt^T = W_out * M -> [B,256,128], K=128
  gemm_nt<<<dim3(IC_ / 16, C_ / 16, B_), 32, 0, stream>>>(
      w_out, 0, IC_, Mt, (long)IC_ * IC_, IC_, T2, (long)C_ * IC_, IC_, IC_, 1.0f);

  // 6. Atil = (T2 * WthT^T) / HW -> [B,256,272], K=128
  gemm_nt<<<dim3(CP_ / 16, C_ / 16, B_), 32, 0, stream>>>(
      T2, (long)C_ * IC_, IC_, wthT, 0, IC_, Atil, (long)C_ * CP_, CP_, IC_,
      1.0f / (float)HW_);

  // 7. fold BatchNorm + b_out into A2
  finalize_A<<<((long)B_ * C_ * CP_ + 255) / 256, 256, 0, stream>>>(
      Atil, b_out, bn_gamma, bn_beta, bn_mean, bn_var, A2);

  // 8. out = x + A2 @ X~
  final_out<<<dim3(HW_ / 32, C_ / 32, B_), 32, 0, stream>>>(x, A2, out);
}